// YoloLayer_18073222381691
// MI455X (gfx1250) — compile-verified
//
#include <hip/hip_runtime.h>

// ---------------- problem constants (fixed by setup_inputs) ----------------
constexpr int BATCH   = 16;
constexpr int NANCH   = 3;
constexpr int NCLS    = 80;
constexpr int CPA     = NCLS + 5;        // 85 channels per anchor
constexpr int DIMW    = 76;
constexpr int DIMH    = 76;
constexpr int HW      = DIMW * DIMH;     // 5776
constexpr int THREADS = 256;             // 8 wave32 per block
constexpr int CHUNK   = 16;              // class channels staged per pipeline stage
constexpr int NCHUNK  = NCLS / CHUNK;    // 5
// LDS: 2 buffers x 8 waves x CHUNK channels x 32 lanes x 4B = 32 KB
constexpr int WAVE_BYTES = CHUNK * 32 * 4;           // 2048 B per wave per buffer
constexpr int BUF_BYTES  = (THREADS / 32) * WAVE_BYTES; // 16384 B per buffer
constexpr int LDS_BYTES  = 2 * BUF_BYTES;            // 32768 B

__device__ __forceinline__ float fast_sigmoid(float x) {
    return __builtin_amdgcn_rcpf(1.0f + __expf(-x));
}

__global__ __launch_bounds__(THREADS)
void yolo_decode_kernel(const float* __restrict__ in,
                        const float* __restrict__ anchors,
                        float* __restrict__ out) {
    extern __shared__ float lds[];

    const int tid  = threadIdx.x;
    const int lane = tid & 31;
    const int wave = tid >> 5;

    constexpr int blocksPerSlice = (HW + THREADS - 1) / THREADS; // 23
    const int slice = blockIdx.x / blocksPerSlice;   // = b*NANCH + a
    const int pblk  = blockIdx.x % blocksPerSlice;
    const int a     = slice % NANCH;

    const int p  = pblk * THREADS + tid;
    const int pl = (p < HW) ? p : (HW - 1);          // clamp: keep EXEC full

    const int inBase = slice * (CPA * HW) + pl;      // fits in int (23.5M elems)
    const float* gclass = in + inBase + 5 * HW;      // first class channel, per lane

    // per-lane LDS byte offset inside a buffer: wave*2048 + lane*4 (+ c*128)
    const unsigned ldsLane = (unsigned)(wave * WAVE_BYTES + lane * 4);

    // -------- async pipeline: issue one chunk of 16 class channels ---------
    auto issue_chunk = [&](int k, int buf) {
#pragma unroll
        for (int c = 0; c < CHUNK; ++c) {
            const float* gp   = gclass + (k * CHUNK + c) * HW;
            unsigned     loff = (unsigned)(buf * BUF_BYTES) + ldsLane + (unsigned)(c * 128);
            asm volatile("global_load_async_to_lds_b32 %0, %1, off"
                         :: "v"(loff), "v"(gp)
                         : "memory");
        }
    };

    issue_chunk(0, 0);   // prefetch first chunk before touching box channels

    // -------- box channels: direct coalesced loads (overlap async latency) --
    const float t0 = in[inBase + 0 * HW];
    const float t1 = in[inBase + 1 * HW];
    const float t2 = in[inBase + 2 * HW];
    const float t3 = in[inBase + 3 * HW];
    const float t4 = in[inBase + 4 * HW];

    const float aw = anchors[2 * a + 0];
    const float ah = anchors[2 * a + 1];

    constexpr float invW = 1.0f / (float)DIMW;
    constexpr float invH = 1.0f / (float)DIMH;
    const float gx = (float)(pl % DIMW);
    const float gy = (float)(pl / DIMW);

    const float xs   = (fast_sigmoid(t0) + gx) * invW;
    const float ys   = (fast_sigmoid(t1) + gy) * invH;
    const float ws   = __expf(t2) * aw * invW;
    const float hs   = __expf(t3) * ah * invH;
    const float conf = fast_sigmoid(t4);

    // -------- online softmax over 80 class logits, staged through LDS ------
    float m    = -__builtin_inff();
    float ssum = 0.0f;
    int   cid  = 0;

    for (int k = 0; k < NCHUNK; ++k) {
        if (k + 1 < NCHUNK) {
            issue_chunk(k + 1, (k + 1) & 1);
            // wait for chunk k (<=16 outstanding from chunk k+1 may remain)
            asm volatile("s_wait_asynccnt 16" ::: "memory");
        } else {
            asm volatile("s_wait_asynccnt 0" ::: "memory");
        }
        const int fbase = ((k & 1) * BUF_BYTES >> 2) + (int)(ldsLane >> 2);
#pragma unroll
        for (int c = 0; c < CHUNK; ++c) {
            const float x   = lds[fbase + c * 32];
            const int   cls = k * CHUNK + c;
            if (x > m) {
                ssum = ssum * __expf(m - x) + 1.0f;  // rescale old sum, add exp(0)
                m    = x;
                cid  = cls;
            } else {
                ssum += __expf(x - m);
            }
        }
    }

    const float cls_prob = __builtin_amdgcn_rcpf(ssum);  // exp(m-m)/sum = 1/sum

    // -------- store 7 outputs per position ---------------------------------
    if (p < HW) {
        const int o = (slice * HW + p) * 7;
        out[o + 0] = xs;
        out[o + 1] = ys;
        out[o + 2] = ws;
        out[o + 3] = hs;
        out[o + 4] = conf;
        out[o + 5] = cls_prob;
        out[o + 6] = (float)cid;
    }
}

extern "C" void kernel_launch(void* const* d_in, const int* in_sizes, int n_in,
                              void* d_out, int out_size, void* d_ws, size_t ws_size,
                              hipStream_t stream) {
    const float* in      = (const float*)d_in[0];
    const float* anchors = (const float*)d_in[1];
    float*       out     = (float*)d_out;

    constexpr int blocksPerSlice = (HW + THREADS - 1) / THREADS;  // 23
    const int grid = BATCH * NANCH * blocksPerSlice;              // 1104 blocks

    yolo_decode_kernel<<<grid, THREADS, LDS_BYTES, stream>>>(in, anchors, out);
}